// SAGERerankerGELU_48885317763294
// MI455X (gfx1250) — compile-verified
//
#include <hip/hip_runtime.h>
#include <hip/hip_bf16.h>

#define DH 128

typedef float v2f __attribute__((ext_vector_type(2)));
typedef float v8f __attribute__((ext_vector_type(8)));

__device__ __forceinline__ float gelu_exact(float x) {
    // torch nn.GELU default: 0.5*x*(1+erf(x/sqrt(2)))
    return 0.5f * x * (1.0f + erff(x * 0.70710678118654752f));
}

// ---------------- zero scratch ----------------
__global__ void zero_f32_kernel(float* __restrict__ p, long long n) {
    long long i = (long long)blockIdx.x * blockDim.x + threadIdx.x;
    long long stride = (long long)gridDim.x * blockDim.x;
    for (; i < n; i += stride) p[i] = 0.0f;
}

// ---------------- pack weights into paired-K layout ----------------
// dst[(k/2)*Ncol + col] = { src[k][col], src[k+1][col] }  -> B fragment = one b64 load
__global__ void pack_pairs_kernel(const float* __restrict__ src, v2f* __restrict__ dst,
                                  int K, int Ncol) {
    int i = blockIdx.x * blockDim.x + threadIdx.x;
    int total = (K >> 1) * Ncol;
    if (i >= total) return;
    int k2 = i / Ncol;
    int col = i - k2 * Ncol;
    v2f v;
    v.x = src[(2 * k2) * Ncol + col];
    v.y = src[(2 * k2 + 1) * Ncol + col];
    dst[i] = v;
}

// ---------------- edge scatter: agg[dst] += h[src]; deg[dst] += 1 ----------------
// one wave32 per edge per iteration; lane moves one float4 (coalesced 512B row);
// prefetch next edge's source row (global_prefetch_b8) to hide random-access latency.
__global__ void scatter_edges_kernel(const float* __restrict__ h,
                                     const long long* __restrict__ src,
                                     const long long* __restrict__ dst,
                                     float* __restrict__ agg,
                                     float* __restrict__ deg,
                                     int E) {
    const int lane = threadIdx.x & 31;
    const int wid  = (blockIdx.x * blockDim.x + threadIdx.x) >> 5;
    const int nw   = (gridDim.x * blockDim.x) >> 5;
    for (int e = wid; e < E; e += nw) {
        int e2 = e + nw;
        if (e2 < E) {
            long long s2 = src[e2];
            if (lane < 4)  // cover the 512B row in 128B lines
                __builtin_prefetch(h + s2 * DH + lane * 32, 0, 3);
        }
        long long s = src[e];
        long long d = dst[e];
        const float4 v = ((const float4*)(h + s * DH))[lane];
        float* ap = agg + d * DH + (lane << 2);
        atomicAdd(ap + 0, v.x);
        atomicAdd(ap + 1, v.y);
        atomicAdd(ap + 2, v.z);
        atomicAdd(ap + 3, v.w);
        if (lane == 0) atomicAdd(deg + d, 1.0f);
    }
}

// ---------------- fused SAGE / projection GEMM via f32 WMMA ----------------
// out[r,c] = post( A1'[r,:]@W1[:,c] (+ A2[r,:]@W2[:,c]) + bias[c] )  (+ add[r,c])
// A1' = A1 / max(deg,1) when USE_MEAN.  256 threads = 8 waves, 16 rows/wave.
// W1p/W2p are paired-K packed [64 x 128] v2f.
template <bool HAS_A2, bool USE_MEAN, bool DO_GELU, bool HAS_ADD>
__global__ void sage_gemm_kernel(const float* __restrict__ A1,
                                 const float* __restrict__ deg,
                                 const v2f* __restrict__ W1p,
                                 const float* __restrict__ A2,
                                 const v2f* __restrict__ W2p,
                                 const float* __restrict__ bias, // [128]
                                 const float* __restrict__ add,
                                 float* __restrict__ out,
                                 int nrows) {
    const int tid  = threadIdx.x;
    const int wave = tid >> 5;
    const int lane = tid & 31;
    const int half = lane >> 4;      // 0: K=k0,k0+1   1: K=k0+2,k0+3
    const int l16  = lane & 15;

    const int m0 = (blockIdx.x * 8 + wave) * 16;

    // A-fragment row for this lane (ISA 7.12.2: 32-bit A 16x4 layout)
    int rowA = m0 + l16;
    int rowC = rowA < nrows ? rowA : (nrows - 1);   // clamp loads, predicate stores

    float inv1 = 1.0f;
    if (USE_MEAN) inv1 = 1.0f / fmaxf(deg[rowC], 1.0f);

    const float* a1p = A1 + (long long)rowC * DH + 2 * half;
    const float* a2p = HAS_A2 ? (A2 + (long long)rowC * DH + 2 * half) : A1;

    v8f acc[8] = {};

    for (int k0 = 0; k0 < DH; k0 += 4) {
        v2f a1 = *(const v2f*)(a1p + k0);      // one b64 load
        a1.x *= inv1;
        a1.y *= inv1;
        v2f a2;
        if (HAS_A2) a2 = *(const v2f*)(a2p + k0);
        const int kp = (k0 >> 1) + half;       // packed row index for K = k0 + 2*half
#pragma unroll
        for (int n = 0; n < 8; ++n) {
            const int col = n * 16 + l16;
            v2f b1 = W1p[kp * DH + col];       // one b64 load, naturally paired
            acc[n] = __builtin_amdgcn_wmma_f32_16x16x4_f32(
                false, a1, false, b1, (short)0, acc[n], false, false);
            if (HAS_A2) {
                v2f b2 = W2p[kp * DH + col];
                acc[n] = __builtin_amdgcn_wmma_f32_16x16x4_f32(
                    false, a2, false, b2, (short)0, acc[n], false, false);
            }
        }
    }

    // C/D layout: VGPR j -> row m0 + j + 8*half, col = n*16 + l16
#pragma unroll
    for (int n = 0; n < 8; ++n) {
        const int col = n * 16 + l16;
        const float bc = bias[col];
#pragma unroll
        for (int j = 0; j < 8; ++j) {
            int r = m0 + half * 8 + j;
            if (r < nrows) {
                float v = acc[n][j] + bc;
                if (DO_GELU) v = gelu_exact(v);
                if (HAS_ADD) v += add[(long long)r * DH + col];
                out[(long long)r * DH + col] = v;
            }
        }
    }
}

// ---------------- score head: gelu(h@Ws1+bs1)@Ws2+bs2, blend with reranker ----------------
// Ws1p is paired-K packed [64 x 64] v2f, staged through LDS.
__global__ void head_kernel(const float* __restrict__ H,
                            const float* __restrict__ rr,
                            const v2f* __restrict__ Ws1p,   // packed [64*64]
                            const float* __restrict__ bs1,  // [64]
                            const float* __restrict__ Ws2,  // [64]
                            const float* __restrict__ bs2,  // [1]
                            const float* __restrict__ alpha_logit,
                            float* __restrict__ out,
                            int nrows) {
    __shared__ v2f   sW[64 * 64];   // 32KB
    __shared__ float sb1[64];
    __shared__ float sW2[64];

    const int tid = threadIdx.x;
    for (int i = tid; i < 64 * 64; i += 256) sW[i] = Ws1p[i];
    if (tid < 64) { sb1[tid] = bs1[tid]; sW2[tid] = Ws2[tid]; }
    __syncthreads();

    const int wave = tid >> 5;
    const int lane = tid & 31;
    const int half = lane >> 4;
    const int l16  = lane & 15;
    const int m0 = (blockIdx.x * 8 + wave) * 16;

    int rowA = m0 + l16;
    int rowC = rowA < nrows ? rowA : (nrows - 1);
    const float* ap = H + (long long)rowC * DH + 2 * half;

    v8f acc[4] = {};
    for (int k0 = 0; k0 < DH; k0 += 4) {
        v2f a = *(const v2f*)(ap + k0);
        const int kp = (k0 >> 1) + half;
#pragma unroll
        for (int n = 0; n < 4; ++n) {
            const int col = n * 16 + l16;
            v2f b = sW[kp * 64 + col];         // one ds_load_b64
            acc[n] = __builtin_amdgcn_wmma_f32_16x16x4_f32(
                false, a, false, b, (short)0, acc[n], false, false);
        }
    }

    const float alpha = 1.0f / (1.0f + expf(-alpha_logit[0]));
    const float b2 = bs2[0];

#pragma unroll
    for (int j = 0; j < 8; ++j) {
        float p = 0.0f;
#pragma unroll
        for (int n = 0; n < 4; ++n) {
            const int col = n * 16 + l16;
            p += gelu_exact(acc[n][j] + sb1[col]) * sW2[col];
        }
        // reduce across the 16 lanes sharing this row (xor of bits 0..3 stays in-half)
        p += __shfl_xor(p, 1, 32);
        p += __shfl_xor(p, 2, 32);
        p += __shfl_xor(p, 4, 32);
        p += __shfl_xor(p, 8, 32);
        int r = m0 + half * 8 + j;
        if (l16 == 0 && r < nrows) {
            float s = p + b2;
            out[r] = alpha * rr[r] + (1.0f - alpha) * s;
        }
    }
}

extern "C" void kernel_launch(void* const* d_in, const int* in_sizes, int n_in,
                              void* d_out, int out_size, void* d_ws, size_t ws_size,
                              hipStream_t stream) {
    const float*     x    = (const float*)d_in[0];
    const long long* ei   = (const long long*)d_in[1];   // int64 [2, E]
    const float*     rr   = (const float*)d_in[2];
    const float*     Wp   = (const float*)d_in[3];
    const float*     bp   = (const float*)d_in[4];
    const float*     Wl0  = (const float*)d_in[5];
    const float*     bl0  = (const float*)d_in[6];
    const float*     Wr0  = (const float*)d_in[7];
    const float*     Wl1  = (const float*)d_in[8];
    const float*     bl1  = (const float*)d_in[9];
    const float*     Wr1  = (const float*)d_in[10];
    const float*     Ws1  = (const float*)d_in[11];
    const float*     bs1  = (const float*)d_in[12];
    const float*     Ws2  = (const float*)d_in[13];
    const float*     bs2  = (const float*)d_in[14];
    const float*     alog = (const float*)d_in[15];

    const int N = in_sizes[0] / DH;          // 100000
    const int E = in_sizes[1] / 2;           // 1600000
    const long long* src = ei;
    const long long* dst = ei + E;

    // workspace layout (floats): R | H | AGG | DEG | packed weights (~155 MB)
    float* R   = (float*)d_ws;
    float* H   = R + (long long)N * DH;
    float* AGG = H + (long long)N * DH;
    float* DEG = AGG + (long long)N * DH;
    v2f* PWp  = (v2f*)(DEG + N);             // 8B-aligned (even float count before it)
    v2f* PWl0 = PWp  + 64 * DH;
    v2f* PWr0 = PWl0 + 64 * DH;
    v2f* PWl1 = PWr0 + 64 * DH;
    v2f* PWr1 = PWl1 + 64 * DH;
    v2f* PWs1 = PWr1 + 64 * DH;

    const int gemm_blocks = (N + 127) / 128;          // 8 waves x 16 rows per block
    const long long zn = (long long)N * DH + N;       // AGG + DEG contiguous
    dim3 blk(256);

    // 0. pack all weights into paired-K layout (one-time, tiny)
    pack_pairs_kernel<<<32, 256, 0, stream>>>(Wp,  PWp,  DH, DH);
    pack_pairs_kernel<<<32, 256, 0, stream>>>(Wl0, PWl0, DH, DH);
    pack_pairs_kernel<<<32, 256, 0, stream>>>(Wr0, PWr0, DH, DH);
    pack_pairs_kernel<<<32, 256, 0, stream>>>(Wl1, PWl1, DH, DH);
    pack_pairs_kernel<<<32, 256, 0, stream>>>(Wr1, PWr1, DH, DH);
    pack_pairs_kernel<<<16, 256, 0, stream>>>(Ws1, PWs1, DH, 64);

    // 1. residual = x @ Wp + bp
    sage_gemm_kernel<false, false, false, false><<<gemm_blocks, blk, 0, stream>>>(
        x, nullptr, PWp, nullptr, nullptr, bp, nullptr, R, N);

    // 2. layer 0 aggregation
    zero_f32_kernel<<<2048, 256, 0, stream>>>(AGG, zn);
    scatter_edges_kernel<<<4096, blk, 0, stream>>>(x, src, dst, AGG, DEG, E);

    // 3. h = gelu(mean@Wl0 + bl0 + x@Wr0) + residual
    sage_gemm_kernel<true, true, true, true><<<gemm_blocks, blk, 0, stream>>>(
        AGG, DEG, PWl0, x, PWr0, bl0, R, H, N);

    // 4. layer 1 aggregation
    zero_f32_kernel<<<2048, 256, 0, stream>>>(AGG, zn);
    scatter_edges_kernel<<<4096, blk, 0, stream>>>(H, src, dst, AGG, DEG, E);

    // 5. h2 = gelu(mean@Wl1 + bl1 + h@Wr1) + h   (overwrite R)
    sage_gemm_kernel<true, true, true, true><<<gemm_blocks, blk, 0, stream>>>(
        AGG, DEG, PWl1, H, PWr1, bl1, H, R, N);

    // 6. score head + blend
    head_kernel<<<gemm_blocks, blk, 0, stream>>>(
        R, rr, PWs1, bs1, Ws2, bs2, alog, (float*)d_out, N);
}